// GlobalGNN_2224793059853
// MI455X (gfx1250) — compile-verified
//
#include <hip/hip_runtime.h>
#include <hip/hip_bf16.h>

#define BN_EPS 1e-5f

typedef float v2f __attribute__((ext_vector_type(2)));
typedef float v8f __attribute__((ext_vector_type(8)));

static inline int cdiv_h(long long a, long long b) { return (int)((a + b - 1) / b); }

// ---------------- utility ----------------
__global__ void k_zero(float* __restrict__ p, int n) {
    int i = blockIdx.x * blockDim.x + threadIdx.x;
    if (i < n) p[i] = 0.0f;
}

// pad x[N,IN] -> xpad[N,KP] with zeros in cols IN..KP-1
__global__ void k_pad_x(const float* __restrict__ x, float* __restrict__ xpad,
                        int n, int IN, int KP) {
    int i = blockIdx.x * blockDim.x + threadIdx.x;
    if (i >= n * KP) return;
    int node = i / KP, f = i - node * KP;
    xpad[i] = (f < IN) ? x[node * IN + f] : 0.0f;
}

// pad W1[IN,F] -> wpad[KP,F] with zero rows IN..KP-1
__global__ void k_pad_w(const float* __restrict__ w, float* __restrict__ wpad,
                        int IN, int KP, int F) {
    int i = blockIdx.x * blockDim.x + threadIdx.x;
    if (i >= KP * F) return;
    int k = i / F;
    wpad[i] = (k < IN) ? w[i] : 0.0f;
}

// deg[dst] += 1 per edge
__global__ void k_degree(const int* __restrict__ dst, float* __restrict__ deg, int E) {
    int e = blockIdx.x * blockDim.x + threadIdx.x;
    if (e < E) atomicAdd(&deg[dst[e]], 1.0f);
}

// in-place: deg -> rsqrt(deg + 1)   (+1 for self loop)
__global__ void k_dinv(float* __restrict__ deg, int n) {
    int i = blockIdx.x * blockDim.x + threadIdx.x;
    if (i < n) deg[i] = rsqrtf(deg[i] + 1.0f);
}

// ---------------- fp32 WMMA GEMM: C[M,FOUT] = A[M,K] @ W[K,FOUT] (+bias) ----------------
// Compile-time K/FOUT => branch-free fully-unrolled v_wmma_f32_16x16x4_f32 chain.
// One wave computes a full 16 x FOUT row stripe: FOUT/16 accumulators share each A fragment.
// Requires: M % 16 == 0 (N=100000 and G=256 both are), K % 4 == 0 (padded).
// A frag: lanes 0-15 -> M=lane, K={k,k+1}; lanes 16-31 -> M=lane-16, K={k+2,k+3} (b64 load).
// B frag: needs (k,n),(k+1,n) adjacent -> stage W in LDS k-pair-interleaved:
//   element (k,n) at (k>>1)*2F + ((2n + (k&1)) ^ (((k>>1)&1)<<5))
// so each fragment is ONE aligned ds_load_b64, and the XOR-32 swizzle puts the two
// half-waves (even/odd k-pair) on disjoint 32-bank groups => conflict-free.
// C/D: VGPR i holds M=i (lanes 0-15) / M=i+8 (lanes 16-31), N=lane&15.
template <int K, int FOUT>
__global__ void k_gemm_wmma(const float* __restrict__ A, const float* __restrict__ W,
                            const float* __restrict__ bias, float* __restrict__ C,
                            int Mtiles) {
    constexpr int S = 2 * FOUT;                 // floats per k-pair row
    __shared__ float sW[K * FOUT];
    for (int i = threadIdx.x; i < K * FOUT; i += blockDim.x) {
        int k = i / FOUT, n = i - k * FOUT;
        int kh = k >> 1;
        sW[kh * S + ((2 * n + (k & 1)) ^ ((kh & 1) << 5))] = W[i];
    }
    __syncthreads();

    const int wave = blockIdx.x * (blockDim.x >> 5) + (threadIdx.x >> 5);
    if (wave >= Mtiles) return;                 // wave-uniform: EXEC all-ones at WMMA
    const int lane  = threadIdx.x & 31;
    const int nlane = lane & 15;
    const int koff  = (lane >> 4) << 1;         // 0 | 2
    const float* arow = A + (size_t)(wave * 16 + nlane) * K + koff;
    const float* brow = &sW[(2 * nlane) ^ ((koff & 2) << 4)];  // ^0 or ^32 per half-wave

    constexpr int NT = FOUT / 16;
    v8f acc[NT] = {};
#pragma unroll
    for (int k0 = 0; k0 < K; k0 += 4) {
        const v2f a = *(const v2f*)(arow + k0);                   // global_load_b64
        const int kh = (k0 + koff) >> 1;
#pragma unroll
        for (int t = 0; t < NT; ++t) {
            const v2f b = *(const v2f*)(brow + kh * S + t * 32);  // ds_load_b64
            acc[t] = __builtin_amdgcn_wmma_f32_16x16x4_f32(false, a, false, b,
                                                           (short)0, acc[t], false, false);
        }
    }

    const int mbase = wave * 16 + ((lane >> 4) << 3);
#pragma unroll
    for (int t = 0; t < NT; ++t) {
        const float bv = bias ? bias[t * 16 + nlane] : 0.0f;
#pragma unroll
        for (int i = 0; i < 8; ++i)
            C[(size_t)(mbase + i) * FOUT + t * 16 + nlane] = acc[t][i] + bv;
    }
}

// ---------------- edge scatter: agg[dst] += hlin[src] * dinv[src]*dinv[dst] ----------------
// one thread per (edge, 4-feature group); float4 gather + 4 float atomics
__global__ void k_edge_agg(const float* __restrict__ hlin, const int* __restrict__ src,
                           const int* __restrict__ dst, const float* __restrict__ dinv,
                           float* __restrict__ agg, int E, int F) {
    const int groups = F >> 2;
    const int t = blockIdx.x * blockDim.x + threadIdx.x;
    if (t >= E * groups) return;
    const int e = t / groups;
    const int g = t - e * groups;
    const int s = src[e];
    const int d = dst[e];
    const float norm = dinv[s] * dinv[d];
    const float4 v = *(const float4*)(hlin + (long long)s * F + (g << 2));
    float* o = agg + (long long)d * F + (g << 2);
    atomicAdd(o + 0, v.x * norm);
    atomicAdd(o + 1, v.y * norm);
    atomicAdd(o + 2, v.z * norm);
    atomicAdd(o + 3, v.w * norm);
}

// ---------------- fused self-loop + bias + BN(eval) + ReLU ----------------
__global__ void k_post_bn_relu(const float* __restrict__ agg, const float* __restrict__ hlin,
                               const float* __restrict__ dinv, const float* __restrict__ b,
                               const float* __restrict__ gamma, const float* __restrict__ beta,
                               const float* __restrict__ mean, const float* __restrict__ var,
                               float* __restrict__ out, int n, int F) {
    int i = blockIdx.x * blockDim.x + threadIdx.x;
    if (i >= n * F) return;
    int node = i / F, f = i - node * F;
    float di = dinv[node];
    float v = agg[i] + hlin[i] * di * di + b[f];
    v = (v - mean[f]) * (gamma[f] * rsqrtf(var[f] + BN_EPS)) + beta[f];
    out[i] = fmaxf(v, 0.0f);
}

// ---------------- fused self-loop + bias + ReLU (layer 3) ----------------
__global__ void k_post_relu(const float* __restrict__ agg, const float* __restrict__ hlin,
                            const float* __restrict__ dinv, const float* __restrict__ b,
                            float* __restrict__ out, int n, int F) {
    int i = blockIdx.x * blockDim.x + threadIdx.x;
    if (i >= n * F) return;
    int node = i / F, f = i - node * F;
    float di = dinv[node];
    out[i] = fmaxf(agg[i] + hlin[i] * di * di + b[f], 0.0f);
}

// ---------------- pooling ----------------
__global__ void k_count(const int* __restrict__ batch, float* __restrict__ counts, int n) {
    int i = blockIdx.x * blockDim.x + threadIdx.x;
    if (i < n) atomicAdd(&counts[batch[i]], 1.0f);
}

// pool layout [G, 2*OUT]: cols 0..OUT-1 = sum, cols OUT..2*OUT-1 = max.
// h3 >= 0 (post-ReLU) so float-max == int-bit-max against a zero-initialized buffer.
__global__ void k_pool(const float* __restrict__ h3, const int* __restrict__ batch,
                       float* __restrict__ pool, int n, int OUT) {
    int i = blockIdx.x * blockDim.x + threadIdx.x;
    if (i >= n * OUT) return;
    int node = i / OUT, f = i - node * OUT;
    int g = batch[node];
    float v = h3[i];
    atomicAdd(&pool[(long long)g * (2 * OUT) + f], v);
    atomicMax((int*)&pool[(long long)g * (2 * OUT) + OUT + f], __float_as_int(v));
}

__global__ void k_pool_fin(float* __restrict__ pool, const float* __restrict__ counts,
                           int G, int OUT) {
    int i = blockIdx.x * blockDim.x + threadIdx.x;
    if (i >= G * OUT) return;
    int g = i / OUT, f = i - g * OUT;
    pool[(long long)g * (2 * OUT) + f] /= fmaxf(counts[g], 1.0f);
}

// ---------------- host orchestration ----------------
extern "C" void kernel_launch(void* const* d_in, const int* in_sizes, int n_in,
                              void* d_out, int out_size, void* d_ws, size_t ws_size,
                              hipStream_t stream) {
    const float* x   = (const float*)d_in[0];
    const int*   ei  = (const int*)  d_in[1];
    const int*   bat = (const int*)  d_in[2];
    const float* W1  = (const float*)d_in[3];
    const float* b1  = (const float*)d_in[4];
    const float* W2  = (const float*)d_in[5];
    const float* b2  = (const float*)d_in[6];
    const float* W3  = (const float*)d_in[7];
    const float* b3  = (const float*)d_in[8];
    const float* g1  = (const float*)d_in[9];
    const float* be1 = (const float*)d_in[10];
    const float* m1  = (const float*)d_in[11];
    const float* v1  = (const float*)d_in[12];
    const float* g2  = (const float*)d_in[13];
    const float* be2 = (const float*)d_in[14];
    const float* m2  = (const float*)d_in[15];
    const float* v2  = (const float*)d_in[16];
    const float* Wp  = (const float*)d_in[17];
    const float* bp  = (const float*)d_in[18];

    const int N   = in_sizes[2];                // 100000 (multiple of 16)
    const int E   = in_sizes[1] / 2;            // 1600000
    const int IN  = in_sizes[0] / N;            // 22
    const int H   = in_sizes[4];                // 64
    const int OUT = in_sizes[8];                // 32
    const int G   = (out_size - N * OUT) / OUT; // 256 (multiple of 16)
    const int KP  = 32;                         // padded K for layer 1

    const int* src = ei;
    const int* dst = ei + E;

    float* ws    = (float*)d_ws;
    float* dinv  = ws;                              // N
    float* hlin  = dinv + N;                        // N*H
    float* agg   = hlin + (size_t)N * H;            // N*H
    float* hcur  = agg  + (size_t)N * H;            // N*H
    float* pool  = hcur + (size_t)N * H;            // G*2*OUT
    float* cnts  = pool + (size_t)G * 2 * OUT;      // G
    float* xpad  = cnts + G;                        // N*KP
    float* w1pad = xpad + (size_t)N * KP;           // KP*H

    float* h_out = (float*)d_out;                   // N*OUT
    float* grepr = h_out + (size_t)N * OUT;         // G*OUT

    const int B = 256;
    const int wavesPerBlock = B / 32;
    auto gemm_blocks = [&](int M) { return cdiv_h(M / 16, wavesPerBlock); };

    // degrees -> dinv; pad layer-1 operands
    k_zero<<<cdiv_h(N, B), B, 0, stream>>>(dinv, N);
    k_degree<<<cdiv_h(E, B), B, 0, stream>>>(dst, dinv, E);
    k_dinv<<<cdiv_h(N, B), B, 0, stream>>>(dinv, N);
    k_pad_x<<<cdiv_h((long long)N * KP, B), B, 0, stream>>>(x, xpad, N, IN, KP);
    k_pad_w<<<cdiv_h((long long)KP * H, B), B, 0, stream>>>(W1, w1pad, IN, KP, H);

    // ---- layer 1: xpad[N,32] @ W1pad -> hlin[N,H]; aggregate; BN1+ReLU -> hcur ----
    k_gemm_wmma<32, 64><<<gemm_blocks(N), B, 0, stream>>>(xpad, w1pad, nullptr, hlin, N / 16);
    k_zero<<<cdiv_h((long long)N * H, B), B, 0, stream>>>(agg, N * H);
    k_edge_agg<<<cdiv_h((long long)E * (H / 4), B), B, 0, stream>>>(hlin, src, dst, dinv, agg, E, H);
    k_post_bn_relu<<<cdiv_h((long long)N * H, B), B, 0, stream>>>(agg, hlin, dinv, b1, g1, be1, m1, v1,
                                                                  hcur, N, H);

    // ---- layer 2: hcur[N,H] @ W2 -> hlin; aggregate; BN2+ReLU -> hcur ----
    k_gemm_wmma<64, 64><<<gemm_blocks(N), B, 0, stream>>>(hcur, W2, nullptr, hlin, N / 16);
    k_zero<<<cdiv_h((long long)N * H, B), B, 0, stream>>>(agg, N * H);
    k_edge_agg<<<cdiv_h((long long)E * (H / 4), B), B, 0, stream>>>(hlin, src, dst, dinv, agg, E, H);
    k_post_bn_relu<<<cdiv_h((long long)N * H, B), B, 0, stream>>>(agg, hlin, dinv, b2, g2, be2, m2, v2,
                                                                  hcur, N, H);

    // ---- layer 3: hcur[N,H] @ W3 -> hlin[N,OUT]; aggregate; ReLU -> d_out ----
    k_gemm_wmma<64, 32><<<gemm_blocks(N), B, 0, stream>>>(hcur, W3, nullptr, hlin, N / 16);
    k_zero<<<cdiv_h((long long)N * OUT, B), B, 0, stream>>>(agg, N * OUT);
    k_edge_agg<<<cdiv_h((long long)E * (OUT / 4), B), B, 0, stream>>>(hlin, src, dst, dinv, agg, E, OUT);
    k_post_relu<<<cdiv_h((long long)N * OUT, B), B, 0, stream>>>(agg, hlin, dinv, b3, h_out, N, OUT);

    // ---- pooling: segment mean (sum/count) + segment max, then [G,64] @ Wp + bp ----
    k_zero<<<cdiv_h((long long)G * 2 * OUT, B), B, 0, stream>>>(pool, G * 2 * OUT);
    k_zero<<<cdiv_h(G, B), B, 0, stream>>>(cnts, G);
    k_count<<<cdiv_h(N, B), B, 0, stream>>>(bat, cnts, N);
    k_pool<<<cdiv_h((long long)N * OUT, B), B, 0, stream>>>(h_out, bat, pool, N, OUT);
    k_pool_fin<<<cdiv_h((long long)G * OUT, B), B, 0, stream>>>(pool, cnts, G, OUT);
    k_gemm_wmma<64, 32><<<gemm_blocks(G), B, 0, stream>>>(pool, Wp, bp, grepr, G / 16);
}